// BurgersSolver_75333726371954
// MI455X (gfx1250) — compile-verified
//
#include <hip/hip_runtime.h>
#include <cstddef>

// Burgers 1D: 64 rows x 512 points, 5000 serial steps, snapshot every 50.
// Design: one wave32 per row; 16 elements per lane held in registers;
// neighbor exchange via ds_bpermute (__shfl) -> no barriers, no LDS in hot loop.
// Input row staged to LDS via gfx1250 async global->LDS copy (ASYNCcnt path).

#define NXF            512
#define OUT_X          257
#define OUT_T          101
#define SNAP_STRIDE    50
#define N_SNAPS        100   // snapshots after the initial one
#define EPL            16    // elements per lane (32 lanes * 16 = 512)

// vector + address-space types for the async builtin
// (param0 printed by clang as "int __vector(4) __device__ *" => AS1 v4i*,
//  LDS side correspondingly AS3 v4i*)
typedef int v4i __attribute__((ext_vector_type(4)));
typedef __attribute__((address_space(1))) v4i gv4i;
typedef __attribute__((address_space(3))) v4i lv4i;

__device__ __forceinline__ float shfl32(float v, int srcLane) {
  return __shfl(v, srcLane, 32);
}

__global__ __launch_bounds__(32)
void burgers_row_kernel(const float* __restrict__ u0, float* __restrict__ out) {
  const int b    = blockIdx.x;   // row index (0..63)
  const int lane = threadIdx.x;  // 0..31 (one wave32 per block)

  __shared__ float row[NXF];

  char* s8 = (char*)(u0 + (size_t)b * NXF);  // drop const for addrspace casts
  char* l8 = (char*)row;

  // ---- stage the 2KB input row into LDS via async global->LDS (gfx1250) ----
  // 32 lanes * 16B * 4 iters = 2048B, tracked on ASYNCcnt
  #pragma unroll
  for (int it = 0; it < 4; ++it) {
    const int off = (it * 32 + lane) * 16;
    __builtin_amdgcn_global_load_async_to_lds_b128(
        (gv4i*)(s8 + off), (lv4i*)(l8 + off), 0, 0);
  }
  __builtin_amdgcn_s_wait_asynccnt(0);
  asm volatile("" ::: "memory");  // keep LDS reads below the wait

  // ---- interpolation onto the 513-grid (keep only j = 0..511) ----
  // u_init[j] = u0[j-1]*(j/512) + u0[j]*((512-j)/512); exact fp32 weights.
  float u[EPL];
  #pragma unroll
  for (int k = 0; k < EPL; ++k) {
    const int   j    = lane * EPL + k;
    const float frac = __fmul_rn((float)(NXF - j), 1.0f / (float)NXF); // (512-j)/512 exact
    const float w0   = 1.0f - frac;                                    // j/512 exact
    const float lft  = (j == 0) ? 0.0f : row[j - 1];
    u[k] = __fadd_rn(__fmul_rn(lft, w0), __fmul_rn(row[j], frac));
  }

  const float c1 = 0.0512f;    // DT/(2*DX)
  const float c2 = 0.262144f;  // NU*DT/DX^2

  float* outb = out + (size_t)b * (OUT_X * OUT_T);

  // snapshot writer: even grid points -> x_out = j/2 (0..255); x_out=256 wraps to u[0]
  auto write_snap = [&](int t) {
    #pragma unroll
    for (int k = 0; k < EPL; k += 2) {
      const int j = lane * EPL + k;
      outb[(j >> 1) * OUT_T + t] = u[k];
    }
    if (lane == 0) outb[256 * OUT_T + t] = u[0];
  };

  write_snap(0);

  for (int snap = 1; snap <= N_SNAPS; ++snap) {
    for (int ss = 0; ss < SNAP_STRIDE; ++ss) {
      // halo exchange: 2 cross-lane permutes, no barriers
      const float lft = shfl32(u[EPL - 1], (lane + 31) & 31);  // u[16*lane - 1] (periodic)
      const float rgt = shfl32(u[0],       (lane + 1)  & 31);  // u[16*lane + 16] (periodic)
      float um1 = lft;
      float cur = u[0];
      #pragma unroll
      for (int k = 0; k < EPL; ++k) {
        const float up1 = (k + 1 < EPL) ? u[k + 1] : rgt;
        const float un  = cur - c1 * (up1 - um1) * cur
                              + c2 * (um1 + up1 - 2.0f * cur);
        um1  = cur;
        cur  = up1;
        u[k] = un;
      }
    }
    write_snap(snap);
  }
}

extern "C" void kernel_launch(void* const* d_in, const int* in_sizes, int n_in,
                              void* d_out, int out_size, void* d_ws, size_t ws_size,
                              hipStream_t stream) {
  (void)n_in; (void)out_size; (void)d_ws; (void)ws_size;
  const float* u0  = (const float*)d_in[0];
  float*       out = (float*)d_out;
  const int batch  = in_sizes[0] / NXF;  // 64 rows
  burgers_row_kernel<<<dim3(batch), dim3(32), 0, stream>>>(u0, out);
}